// InvRT_41085657153637
// MI455X (gfx1250) — compile-verified
//
#include <hip/hip_runtime.h>

// Problem constants (from reference): z[M,N,B,D], eta[N,4], eta_fault[18,4], Mask[M,D]
#define MDIM 4
#define NDIM 64
#define BDIM 512
#define DDIM 512
#define BCHUNK 64                 // b-rows per block
#define NBCHUNK (BDIM / BCHUNK)   // 8
#define THREADS 256               // 8 wave32s
#define UNROLL 4

// Clang native vector: required by __builtin_nontemporal_* and gives b128 vmem ops
typedef float v4f __attribute__((ext_vector_type(4)));

__device__ __forceinline__ float hw_tanh(float x) {
    float t;
    // gfx1250 hardware tanh (TRANS pipe). v_nop inside the asm satisfies the
    // "1 op after TRANS before result use" hazard the compiler can't see.
    asm("v_tanh_f32 %0, %1\n\tv_nop" : "=v"(t) : "v"(x));
    return t;
}

// out[m,n,b,d] = -(e0 + e1 * tanh((z - e2) * e3)),  e* selected per (m,n,d)
__global__ __launch_bounds__(THREADS) void invrt_tanh_kernel(
    const float* __restrict__ z,
    const float* __restrict__ eta,        // [N,4]
    const float* __restrict__ eta_fault,  // [18,4]
    const int*   __restrict__ mask,       // [M,D]
    float* __restrict__ out)
{
    // SoA coefficient rows for this (m,n): 4 * 512 * 4B = 8 KB LDS
    __shared__ __align__(16) float e0s[DDIM];
    __shared__ __align__(16) float e1s[DDIM];
    __shared__ __align__(16) float e2s[DDIM];
    __shared__ __align__(16) float e3s[DDIM];

    const int blk = blockIdx.x;
    const int bc  = blk % NBCHUNK;         // which b-chunk
    const int mn  = blk / NBCHUNK;
    const int n   = mn % NDIM;
    const int m   = mn / NDIM;

    // Build eta_sel[m,n,:] into LDS (2 entries per thread). Tables are tiny
    // and L2-resident; these loads use the default (cached) policy.
    for (int d = threadIdx.x; d < DDIM; d += THREADS) {
        const int    msk = mask[m * DDIM + d];
        const float* src = (msk == 0) ? (eta + 4 * n)
                                      : (eta_fault + 4 * (msk - 1));
        e0s[d] = src[0];
        e1s[d] = src[1];
        e2s[d] = src[2];
        e3s[d] = src[3];
    }
    __syncthreads();

    // Key invariant: group stride (256) is a multiple of DDIM/4 (128), so each
    // thread's d-position never changes. Hoist the 4 coefficient vectors into
    // registers once — the steady-state loop touches no LDS at all.
    const int d4 = threadIdx.x & (DDIM / 4 - 1);
    const v4f a0 = *(const v4f*)&e0s[d4 * 4];
    const v4f a1 = *(const v4f*)&e1s[d4 * 4];
    const v4f a2 = *(const v4f*)&e2s[d4 * 4];
    const v4f a3 = *(const v4f*)&e3s[d4 * 4];

    // Contiguous slab for this block: BCHUNK*DDIM floats
    const size_t base = (((size_t)m * NDIM + n) * BDIM + (size_t)bc * BCHUNK) * DDIM;
    const v4f* __restrict__ zin = (const v4f*)(z + base);
    v4f* __restrict__ zout      = (v4f*)(out + base);

    const int groups = BCHUNK * (DDIM / 4);          // 8192 v4f groups
    const int iters  = groups / (THREADS * UNROLL);  // 8 outer iterations

    int g = threadIdx.x;
    for (int it = 0; it < iters; ++it, g += THREADS * UNROLL) {
        // Issue all UNROLL non-temporal b128 loads up front (4 in flight).
        v4f zv[UNROLL];
        #pragma unroll
        for (int u = 0; u < UNROLL; ++u)
            zv[u] = __builtin_nontemporal_load(&zin[g + u * THREADS]);

        #pragma unroll
        for (int u = 0; u < UNROLL; ++u) {
            v4f r;
            r.x = -__builtin_fmaf(a1.x, hw_tanh((zv[u].x - a2.x) * a3.x), a0.x);
            r.y = -__builtin_fmaf(a1.y, hw_tanh((zv[u].y - a2.y) * a3.y), a0.y);
            r.z = -__builtin_fmaf(a1.z, hw_tanh((zv[u].z - a2.z) * a3.z), a0.z);
            r.w = -__builtin_fmaf(a1.w, hw_tanh((zv[u].w - a2.w) * a3.w), a0.w);
            __builtin_nontemporal_store(r, &zout[g + u * THREADS]);
        }
    }
}

extern "C" void kernel_launch(void* const* d_in, const int* in_sizes, int n_in,
                              void* d_out, int out_size, void* d_ws, size_t ws_size,
                              hipStream_t stream) {
    (void)in_sizes; (void)n_in; (void)out_size; (void)d_ws; (void)ws_size;

    const float* z         = (const float*)d_in[0];
    const float* eta       = (const float*)d_in[1];
    const float* eta_fault = (const float*)d_in[2];
    const int*   mask      = (const int*)d_in[3];
    float*       out       = (float*)d_out;

    const int grid = MDIM * NDIM * NBCHUNK;   // 4*64*8 = 2048 blocks
    invrt_tanh_kernel<<<grid, THREADS, 0, stream>>>(z, eta, eta_fault, mask, out);
}